// SymmetricContraction_31267361915222
// MI455X (gfx1250) — compile-verified
//
#include <hip/hip_runtime.h>
#include <hip/hip_bf16.h>

// ---------------------------------------------------------------------------
// Symmetric contraction (MACE-style), fully factorized:
//   out[b,c,(p)] = sum_e y[b,e] * sum_t G[c][(e,p)][t] * m[b,c][t]
// m = symmetric monomials of x[b,c,0:9]  (165 cubic + 45 quadratic + 9 linear)
// G = per-channel fold of symmetrized U with W.
// Main GEMM: M=524288, K=224, N=48 in fp32 via V_WMMA_F32_16X16X4_F32.
// ---------------------------------------------------------------------------

#define NB   2048
#define NC   256
#define NI   9
#define NE   10
#define T3   165
#define T2   45
#define K3S  23
#define K2S  3
#define K3V  30
#define K2V  4
#define KTOT 224      // padded K (219 real rows)
#define KT   56       // K tiles of 4
#define CTN  3        // N tiles of 16 (48 cols, 40 real)
#define AST  226      // LDS A-panel row stride (floats), conflict-free
#define ZST  49       // LDS Z-panel row stride

// ws layout (float offsets)
#define OFF_U3S 0
#define OFF_U2S (OFF_U3S + T3*K3S)            // 3795
#define OFF_U3V (OFF_U2S + T2*K2S)            // 3930
#define OFF_U2V (OFF_U3V + 3*T3*K3V)          // 18780
#define OFF_GFRAG 19456
#define GFRAG_PER_C (CTN*KT*64)               // 10752 floats per channel

typedef float v2f __attribute__((ext_vector_type(2)));
typedef float v8f __attribute__((ext_vector_type(8)));

// ---------------------------------------------------------------------------
__device__ inline void t3_decode(int t, int& oa, int& ob, int& oc) {
  int idx = 0;
  for (int a = 0; a < NI; a++)
    for (int b = a; b < NI; b++)
      for (int c = b; c < NI; c++) {
        if (idx == t) { oa = a; ob = b; oc = c; return; }
        idx++;
      }
}
__device__ inline void t2_decode(int t, int& oa, int& ob) {
  int idx = 0;
  for (int a = 0; a < NI; a++)
    for (int b = a; b < NI; b++) {
      if (idx == t) { oa = a; ob = b; return; }
      idx++;
    }
}

template <typename F>
__device__ inline float symsum3(int a, int b, int c, F fetch) {
  const int P[6][3] = {{a,b,c},{a,c,b},{b,a,c},{b,c,a},{c,a,b},{c,b,a}};
  float s = 0.f;
  for (int p = 0; p < 6; p++) {
    int code = P[p][0]*81 + P[p][1]*9 + P[p][2];
    bool dup = false;
    for (int q = 0; q < p; q++) {
      if (P[q][0]*81 + P[q][1]*9 + P[q][2] == code) { dup = true; break; }
    }
    if (!dup) s += fetch(P[p][0], P[p][1], P[p][2]);
  }
  return s;
}

// ---------------------------------------------------------------------------
// Kernel 1: symmetrize U2/U3 (scalar + 3 vector components) into ws.
__global__ void prep_sym(const float* __restrict__ U2_s,
                         const float* __restrict__ U3_s,
                         const float* __restrict__ U2_v,
                         const float* __restrict__ U3_v,
                         float* __restrict__ ws) {
  int id = blockIdx.x * blockDim.x + threadIdx.x;
  const int n3s = T3*K3S, n2s = T2*K2S, n3v = 3*T3*K3V, n2v = 3*T2*K2V;

  if (id < n3s) {
    int t = id / K3S, k = id % K3S;
    int a, b, c; t3_decode(t, a, b, c);
    ws[OFF_U3S + id] = symsum3(a, b, c, [&](int i, int j, int l) {
      return U3_s[((i*NI + j)*NI + l)*K3S + k];
    });
    return;
  }
  id -= n3s;
  if (id < n2s) {
    int t = id / K2S, k = id % K2S;
    int a, b; t2_decode(t, a, b);
    float s = U2_s[(a*NI + b)*K2S + k];
    if (a != b) s += U2_s[(b*NI + a)*K2S + k];
    ws[OFF_U2S + id] = s;
    return;
  }
  id -= n2s;
  if (id < n3v) {
    int v = id / (T3*K3V), r = id % (T3*K3V);
    int t = r / K3V, k = r % K3V;
    int a, b, c; t3_decode(t, a, b, c);
    ws[OFF_U3V + id] = symsum3(a, b, c, [&](int i, int j, int l) {
      return U3_v[(((v*NI + i)*NI + j)*NI + l)*K3V + k];
    });
    return;
  }
  id -= n3v;
  if (id < n2v) {
    int v = id / (T2*K2V), r = id % (T2*K2V);
    int t = r / K2V, k = r % K2V;
    int a, b; t2_decode(t, a, b);
    float s = U2_v[((v*NI + a)*NI + b)*K2V + k];
    if (a != b) s += U2_v[((v*NI + b)*NI + a)*K2V + k];
    ws[OFF_U2V + id] = s;
  }
}

// ---------------------------------------------------------------------------
// Kernel 2: build G[c] (KTOT x 48) directly in WMMA B-fragment layout.
// Fragment slot layout: gfrag[((c*CTN + ct)*KT + kt)*64 + lane*2 + g]
//   holds G[t = kt*4 + g + (lane>=16?2:0)][n = ct*16 + (lane&15)]
// Column n = e*4 + p, p: 0=scalar, 1..3=vector component.
__global__ void prep_gfrag(const float* __restrict__ U1_s,
                           const float* __restrict__ U1_v,
                           const float* __restrict__ W1_s,
                           const float* __restrict__ W2_s,
                           const float* __restrict__ W3_s,
                           const float* __restrict__ W1_v,
                           const float* __restrict__ W2_v,
                           const float* __restrict__ W3_v,
                           const float* __restrict__ ws_u,
                           float* __restrict__ gfrag) {
  long id = (long)blockIdx.x * blockDim.x + threadIdx.x;
  if (id >= (long)NC * GFRAG_PER_C) return;
  int c    = (int)(id / GFRAG_PER_C);
  int rem  = (int)(id % GFRAG_PER_C);
  int ct   = rem / (KT*64); rem %= (KT*64);
  int kt   = rem / 64;
  int slot = rem % 64;
  int lane = slot >> 1, g = slot & 1;
  int t = kt*4 + g + ((lane >= 16) ? 2 : 0);
  int n = ct*16 + (lane & 15);

  float val = 0.f;
  if (n < 40 && t < 219) {
    int e = n >> 2, p = n & 3;
    if (t < T3) {                       // cubic block
      if (p == 0) {
        const float* u = ws_u + OFF_U3S + t*K3S;
        const float* w = W3_s + e*K3S*NC + c;
        float s = 0.f;
        for (int k = 0; k < K3S; k++) s += u[k] * w[k*NC];
        val = s;
      } else {
        int v = p - 1;
        const float* u = ws_u + OFF_U3V + (v*T3 + t)*K3V;
        const float* w = W3_v + e*K3V*NC + c;
        float s = 0.f;
        for (int k = 0; k < K3V; k++) s += u[k] * w[k*NC];
        val = s;
      }
    } else if (t < T3 + T2) {           // quadratic block
      int t2 = t - T3;
      if (p == 0) {
        const float* u = ws_u + OFF_U2S + t2*K2S;
        const float* w = W2_s + e*K2S*NC + c;
        float s = 0.f;
        for (int k = 0; k < K2S; k++) s += u[k] * w[k*NC];
        val = s;
      } else {
        int v = p - 1;
        const float* u = ws_u + OFF_U2V + (v*T2 + t2)*K2V;
        const float* w = W2_v + e*K2V*NC + c;
        float s = 0.f;
        for (int k = 0; k < K2V; k++) s += u[k] * w[k*NC];
        val = s;
      }
    } else {                            // linear block
      int i = t - (T3 + T2);
      if (p == 0) val = U1_s[i]          * W1_s[e*NC + c];
      else        val = U1_v[(p-1)*NI+i] * W1_v[e*NC + c];
    }
  }
  gfrag[id] = val;
}

// ---------------------------------------------------------------------------
// Kernel 3: main WMMA GEMM + epilogue.
// grid = (16 b-chunks, 256 channels), block = 256 threads = 8 waves.
// Wave w owns rows [w*16, w*16+16) of the 128-row chunk.
__global__ void __launch_bounds__(256)
sc_main(const float* __restrict__ x, const float* __restrict__ y,
        const float* __restrict__ gfrag, float* __restrict__ out) {
  extern __shared__ float lds[];
  float* Gl = lds;                      // 10752 floats (B fragments for this c)
  float* Ab = Gl + GFRAG_PER_C;         // 128 rows * AST = 28928 floats
  float* Zb = Ab + 128*AST;             // 128 rows * ZST = 6272 floats

  const int tid  = threadIdx.x;
  const int wave = tid >> 5;
  const int lane = tid & 31;
  const int c    = blockIdx.y;
  const int b0   = blockIdx.x * 128;

  // --- stage G[c] fragments to LDS (coalesced) ---
  {
    const float* gsrc = gfrag + (size_t)c * GFRAG_PER_C;
    for (int i = tid; i < GFRAG_PER_C; i += 256) Gl[i] = gsrc[i];
  }

  // --- build monomial rows (lanes 0..15 of each wave: one row each) ---
  if (lane < 16) {
    const int r = wave*16 + lane;             // 0..127
    const int b = b0 + r;
    const float* xp = x + ((size_t)b*NC + c)*NI;
    float xv[NI];
#pragma unroll
    for (int i = 0; i < NI; i++) xv[i] = xp[i];
    float* arow = Ab + r*AST;
    int t = 0;
    for (int a = 0; a < NI; a++)
      for (int bq = a; bq < NI; bq++) {
        float p2 = xv[a]*xv[bq];
        for (int cq = bq; cq < NI; cq++) arow[t++] = p2*xv[cq];   // 165
      }
    for (int a = 0; a < NI; a++)
      for (int bq = a; bq < NI; bq++) arow[t++] = xv[a]*xv[bq];   // 45
#pragma unroll
    for (int i = 0; i < NI; i++) arow[t++] = xv[i];               // 9
    for (; t < KTOT; t++) arow[t] = 0.f;                          // pad
  }
  __syncthreads();

  // --- GEMM: 16x48 tile per wave, K=224, v_wmma_f32_16x16x4_f32 ---
  {
    const int half = lane >> 4;                  // selects K pair {0,1} vs {2,3}
    const int l16  = lane & 15;
    const float* arow = Ab + (wave*16 + l16)*AST + half*2;
    const float* g0 = Gl + 0*KT*64 + lane*2;
    const float* g1 = Gl + 1*KT*64 + lane*2;
    const float* g2 = Gl + 2*KT*64 + lane*2;

    v8f acc0 = {0,0,0,0,0,0,0,0};
    v8f acc1 = {0,0,0,0,0,0,0,0};
    v8f acc2 = {0,0,0,0,0,0,0,0};

#pragma unroll 8
    for (int kt = 0; kt < KT; kt++) {
      v2f A  = *(const v2f*)(arow + kt*4);
      v2f B0 = *(const v2f*)(g0 + kt*64);
      v2f B1 = *(const v2f*)(g1 + kt*64);
      v2f B2 = *(const v2f*)(g2 + kt*64);
      acc0 = __builtin_amdgcn_wmma_f32_16x16x4_f32(false, A, false, B0,
                                                   (short)0, acc0, false, false);
      acc1 = __builtin_amdgcn_wmma_f32_16x16x4_f32(false, A, false, B1,
                                                   (short)0, acc1, false, false);
      acc2 = __builtin_amdgcn_wmma_f32_16x16x4_f32(false, A, false, B2,
                                                   (short)0, acc2, false, false);
    }

    // C/D layout: vgpr g, lanes 0-15 -> row g ; lanes 16-31 -> row g+8
    const int zr = wave*16 + (half ? 8 : 0);
#pragma unroll
    for (int g = 0; g < 8; g++) {
      Zb[(zr+g)*ZST +  0 + l16] = acc0[g];
      Zb[(zr+g)*ZST + 16 + l16] = acc1[g];
      Zb[(zr+g)*ZST + 32 + l16] = acc2[g];
    }
  }
  __syncthreads();

  // --- epilogue: out[b,c,p] = sum_e y[b,e] * Z[r][e*4+p] ---
  {
    const int r  = tid >> 1;          // 0..127
    const int p0 = (tid & 1)*2;       // {0,1} or {2,3}
    const int b  = b0 + r;
    const float* yb = y + (size_t)b*NE;
    float yv[NE];
#pragma unroll
    for (int e = 0; e < NE; e++) yv[e] = yb[e];
#pragma unroll
    for (int pp = 0; pp < 2; pp++) {
      const int p = p0 + pp;
      float s = 0.f;
#pragma unroll
      for (int e = 0; e < NE; e++) s += yv[e] * Zb[r*ZST + e*4 + p];
      // output layout: [b, 0:256]=scalar, [b, 256 + c*3 + v]=vector
      size_t o = (size_t)b*(4*NC) +
                 ((p == 0) ? (size_t)c : (size_t)(NC + c*3 + (p-1)));
      out[o] = s;
    }
  }
}

// ---------------------------------------------------------------------------
extern "C" void kernel_launch(void* const* d_in, const int* in_sizes, int n_in,
                              void* d_out, int out_size, void* d_ws, size_t ws_size,
                              hipStream_t stream) {
  const float* x    = (const float*)d_in[0];
  const float* yv   = (const float*)d_in[1];
  const float* U1_s = (const float*)d_in[2];
  const float* U2_s = (const float*)d_in[3];
  const float* U3_s = (const float*)d_in[4];
  const float* U1_v = (const float*)d_in[5];
  const float* U2_v = (const float*)d_in[6];
  const float* U3_v = (const float*)d_in[7];
  const float* W1_s = (const float*)d_in[8];
  const float* W2_s = (const float*)d_in[9];
  const float* W3_s = (const float*)d_in[10];
  const float* W1_v = (const float*)d_in[11];
  const float* W2_v = (const float*)d_in[12];
  const float* W3_v = (const float*)d_in[13];
  float* ws  = (float*)d_ws;              // uses ~11.1 MB of scratch
  float* outp = (float*)d_out;

  // 1) symmetrize U tensors
  {
    const int n = T3*K3S + T2*K2S + 3*T3*K3V + 3*T2*K2V;   // 19320
    prep_sym<<<(n + 255)/256, 256, 0, stream>>>(U2_s, U3_s, U2_v, U3_v, ws);
  }
  // 2) per-channel G in B-fragment layout
  {
    const long n = (long)NC * GFRAG_PER_C;                 // 2752512
    prep_gfrag<<<(unsigned)((n + 255)/256), 256, 0, stream>>>(
        U1_s, U1_v, W1_s, W2_s, W3_s, W1_v, W2_v, W3_v, ws, ws + OFF_GFRAG);
  }
  // 3) main WMMA kernel
  {
    const size_t smem = (size_t)(GFRAG_PER_C + 128*AST + 128*ZST) * sizeof(float);
    sc_main<<<dim3(NB/128, NC), 256, smem, stream>>>(x, yv, ws + OFF_GFRAG, outp);
  }
}